// MAD_4612794876398
// MI455X (gfx1250) — compile-verified
//
#include <hip/hip_runtime.h>
#include <math.h>

// ---------------------------------------------------------------------------
// Problem constants: n=1024 query images, S=8 refs each.
// LeNet: 3x32x32 -> conv5(6) -> pool2 -> conv5(16) -> pool2 -> 400 feats.
// H=12 hidden, C=10 classes, K=400 feature dim.
// All matrix math runs on fp32 WMMA (v_wmma_f32_16x16x4_f32) — full reference
// precision while using the CDNA5 matrix pipe.
// ---------------------------------------------------------------------------
#define NIMG   1024
#define SREF   8
#define NTOT   (NIMG + NIMG * SREF)   // 9216 images total
#define FDIM   400
#define HID    12
#define NCLS   10
#define IMSZ   (3 * 32 * 32)          // 3072

// LDS geometry for the fused LeNet kernel
#define SIMG_PAD   892                // zero region for K-padding of conv1 im2col
#define SIMG_SZ    (IMSZ + SIMG_PAD)  // 3964
#define K1         75                 // conv1 K = 3*5*5
#define K1P        76                 // padded to multiple of 4
#define SW1_LD     77                 // LDS row stride (odd mod 64 -> no bank conflict)
#define P1_SZ      (6 * 14 * 14)      // 1176
#define P1_PAD     196                // zero region for K-padding of conv2 im2col
#define K2         150                // conv2 K = 6*5*5
#define K2P        152
#define SW2_LD     153

typedef __attribute__((ext_vector_type(2))) float v2f;
typedef __attribute__((ext_vector_type(8))) float v8f;
typedef __attribute__((__vector_size__(16))) int v4i_vec;   // matches builtin param

#if __has_builtin(__builtin_amdgcn_global_load_async_to_lds_b128) && \
    __has_builtin(__builtin_amdgcn_s_wait_asynccnt)
#define HAVE_ASYNC_LDS 1
#endif

static __device__ __forceinline__ v8f wmma_f32x4(v2f a, v2f b, v8f c) {
    // 8 args: (neg_a, A, neg_b, B, c_mod, C, reuse_a, reuse_b)
    return __builtin_amdgcn_wmma_f32_16x16x4_f32(false, a, false, b,
                                                 (short)0, c, false, false);
}

// ---------------------------------------------------------------------------
// Kernel 1: fused LeNet feature extractor, one workgroup (8 waves) per image.
// Both convolutions are implicit-GEMM on fp32 WMMA. im2col address trick:
//   patch_offset(k, pixel) = kOff[k] + pixOff[pixel]
// with kOff a tiny LDS table; K-pad taps point into a zeroed LDS region so the
// WMMA K-loop is branchless with EXEC all ones (required by WMMA).
//
// Fragment layouts (CDNA5 ISA 7.12.2, 32-bit):
//   A 16x4 : row m = lane&15; VGPR0/1 hold K = kh / kh+1 with kh = 2*(lane>>4)
//   B 4x16 : col n = lane&15; same kh split (mirror of A)
//   C/D    : VGPR i -> row i + 8*(lane>>4), col = lane&15
//
// Image staging (the bandwidth floor of the whole problem) uses the gfx1250
// async global->LDS path (ASYNCcnt) when available, else b128 copies.
// ---------------------------------------------------------------------------
__global__ void __launch_bounds__(256)
lenet_features(const float* __restrict__ img, const float* __restrict__ train,
               const int* __restrict__ refs,
               const float* __restrict__ w1, const float* __restrict__ b1,
               const float* __restrict__ w2, const float* __restrict__ b2,
               float* __restrict__ feat /* NTOT x FDIM */)
{
    __shared__ __align__(16) float simg[SIMG_SZ];   // image + zero pad
    __shared__ float sw1p[16 * SW1_LD];      // conv1 weights, 16x76 zero-padded
    __shared__ float sw2p[16 * SW2_LD];      // conv2 weights, 16x152 zero-padded
    __shared__ float sb1[6];
    __shared__ float sb2[16];
    __shared__ int   kOff1[K1P];
    __shared__ int   kOff2[K2P];
    __shared__ float c1[6 * 28 * 28];        // conv1+relu (4704)
    __shared__ float p1[P1_SZ + P1_PAD];     // pool1 + zero pad (1372)
    __shared__ float c2[16 * 10 * 10];       // conv2+relu (1600)

    const int b  = blockIdx.x;
    const int t  = threadIdx.x;
    const int nt = blockDim.x;               // 256
    const int lane = t & 31;
    const int wave = t >> 5;
    const int m  = lane & 15;                // A row / B col within a tile
    const int kh = (lane >> 4) << 1;         // K sub-offset within a step: 0 or 2

    // ---- stage image ----
    const float* src;
    if (b < NIMG) {
        src = img + (size_t)b * IMSZ;
    } else {
        src = train + (size_t)refs[b - NIMG] * IMSZ;   // refs flattened (1024*8,)
    }
#ifdef HAVE_ASYNC_LDS
    {
        __attribute__((address_space(1))) v4i_vec* gsrc =
            (__attribute__((address_space(1))) v4i_vec*)src;
        __attribute__((address_space(3))) v4i_vec* ldst =
            (__attribute__((address_space(3))) v4i_vec*)simg;
        for (int i = t; i < IMSZ / 4; i += nt)   // 768 x b128 chunks
            __builtin_amdgcn_global_load_async_to_lds_b128(gsrc + i, ldst + i, 0, 0);
    }
#else
    {
        const float4* s4 = (const float4*)src;
        float4* d4 = (float4*)simg;
        for (int i = t; i < IMSZ / 4; i += nt) d4[i] = s4[i];
    }
#endif
    for (int i = t; i < SIMG_PAD; i += nt) simg[IMSZ + i] = 0.0f;
    for (int i = t; i < P1_PAD; i += nt) p1[P1_SZ + i] = 0.0f;

    // ---- stage zero-padded weights + im2col tap tables ----
    for (int i = t; i < 16 * SW1_LD; i += nt) {
        const int mm = i / SW1_LD, k = i % SW1_LD;
        sw1p[i] = (mm < 6 && k < K1) ? w1[mm * K1 + k] : 0.0f;
    }
    for (int i = t; i < 16 * SW2_LD; i += nt) {
        const int mm = i / SW2_LD, k = i % SW2_LD;
        sw2p[i] = (k < K2) ? w2[mm * K2 + k] : 0.0f;
    }
    for (int i = t; i < K1P; i += nt) {
        // k -> (ic, ky, kx); offset ic*1024 + ky*32 + kx; pad -> zero region
        kOff1[i] = (i < K1) ? ((i / 25) * 1024 + ((i % 25) / 5) * 32 + (i % 5))
                            : IMSZ;
    }
    for (int i = t; i < K2P; i += nt) {
        kOff2[i] = (i < K2) ? ((i / 25) * 196 + ((i % 25) / 5) * 14 + (i % 5))
                            : P1_SZ;
    }
    if (t < 6)  sb1[t] = b1[t];
    if (t < 16) sb2[t] = b2[t];
#ifdef HAVE_ASYNC_LDS
    __builtin_amdgcn_s_wait_asynccnt(0);     // my async LDS writes landed
#endif
    __syncthreads();

    // ---- conv1 as WMMA implicit GEMM: M=16(6 used), N=784 (49 tiles), K=76 ----
    for (int tile = wave; tile < 49; tile += 8) {
        const int pix = tile * 16 + (lane & 15);               // < 784
        const int pixOff = (pix / 28) * 32 + (pix % 28);
        v8f acc = {0.f, 0.f, 0.f, 0.f, 0.f, 0.f, 0.f, 0.f};
        for (int k0 = 0; k0 < K1P; k0 += 4) {
            v2f a, bb;
            a.x  = sw1p[m * SW1_LD + k0 + kh];
            a.y  = sw1p[m * SW1_LD + k0 + kh + 1];
            bb.x = simg[kOff1[k0 + kh] + pixOff];
            bb.y = simg[kOff1[k0 + kh + 1] + pixOff];
            acc  = wmma_f32x4(a, bb, acc);
        }
        if (lane < 16) {                                       // rows 0..5 = oc 0..5
            #pragma unroll
            for (int i = 0; i < 6; ++i)
                c1[i * 784 + pix] = fmaxf(acc[i] + sb1[i], 0.0f);
        }
    }
    __syncthreads();

    // ---- maxpool 2x2 : (6,14,14) ----
    for (int i = t; i < 6 * 14 * 14; i += nt) {
        const int oc = i / 196, p = i % 196, oy = p / 14, ox = p % 14;
        const float* base = &c1[(oc * 28 + 2 * oy) * 28 + 2 * ox];
        p1[i] = fmaxf(fmaxf(base[0], base[1]), fmaxf(base[28], base[29]));
    }
    __syncthreads();

    // ---- conv2 as WMMA implicit GEMM: M=16, N=112 (100 valid), K=152 ----
    if (wave < 7) {
        const int pix  = wave * 16 + (lane & 15);              // up to 111
        const int pixc = pix < 100 ? pix : 99;                 // clamp pad cols
        const int pixOff = (pixc / 10) * 14 + (pixc % 10);
        v8f acc = {0.f, 0.f, 0.f, 0.f, 0.f, 0.f, 0.f, 0.f};
        for (int k0 = 0; k0 < K2P; k0 += 4) {
            v2f a, bb;
            a.x  = sw2p[m * SW2_LD + k0 + kh];
            a.y  = sw2p[m * SW2_LD + k0 + kh + 1];
            bb.x = p1[kOff2[k0 + kh] + pixOff];
            bb.y = p1[kOff2[k0 + kh + 1] + pixOff];
            acc  = wmma_f32x4(a, bb, acc);
        }
        if (pix < 100) {
            #pragma unroll
            for (int i = 0; i < 8; ++i) {
                const int oc = i + ((lane >> 4) << 3);
                c2[oc * 100 + pix] = fmaxf(acc[i] + sb2[oc], 0.0f);
            }
        }
    }
    __syncthreads();

    // ---- maxpool 2x2 -> (16,5,5)=400 features, straight to global ----
    float* frow = feat + (size_t)b * FDIM;
    for (int i = t; i < FDIM; i += nt) {
        const int oc = i / 25, p = i % 25, oy = p / 5, ox = p % 5;
        const float* base = &c2[(oc * 10 + 2 * oy) * 10 + 2 * ox];
        frow[i] = fmaxf(fmaxf(base[0], base[1]), fmaxf(base[10], base[11]));
    }
}

// ---------------------------------------------------------------------------
// Prep: zero-pad weight rows (Nreal x K) -> (Npad x K) so the GEMM B-fragment
// loads are unconditional (no saveexec-per-load).
// ---------------------------------------------------------------------------
__global__ void __launch_bounds__(256)
pad_weights(const float* __restrict__ W, float* __restrict__ Wp,
            int Nreal, int K)
{
    const int row = blockIdx.x;
    const bool v = row < Nreal;
    const float* srcRow = W + (size_t)(v ? row : 0) * K;
    float* dstRow = Wp + (size_t)row * K;
    for (int k = threadIdx.x; k < K; k += blockDim.x)
        dstRow[k] = v ? srcRow[k] : 0.0f;
}

// ---------------------------------------------------------------------------
// Kernel 2: out[M x ldOut] = A[M x K] * Wp^T + bias, Wp is (ldOut x K) padded.
// One wave per 16x16 tile; branchless float2 (b64) loads for both fragments.
// M multiple of 16 (9216, 1024). Pad cols: acc is exactly 0 -> stored as 0.
// ---------------------------------------------------------------------------
__global__ void __launch_bounds__(256)
wmma_gemm_bias(const float* __restrict__ A, const float* __restrict__ Wp,
               const float* __restrict__ bias, float* __restrict__ out,
               int M, int K, int Nreal, int ldOut, int tilesN)
{
    const int wid  = (blockIdx.x * blockDim.x + threadIdx.x) >> 5;
    const int lane = threadIdx.x & 31;
    const int totalTiles = (M >> 4) * tilesN;
    if (wid >= totalTiles) return;

    const int rt = wid / tilesN, ct = wid % tilesN;
    const int row0 = rt << 4, col0 = ct << 4;

    const int m  = lane & 15;
    const int kh = (lane >> 4) << 1;

    const float* __restrict__ arow = A  + (size_t)(row0 + m) * K + kh;
    const float* __restrict__ wrow = Wp + (size_t)(col0 + m) * K + kh;

    v8f acc = {0.f, 0.f, 0.f, 0.f, 0.f, 0.f, 0.f, 0.f};
    for (int k0 = 0; k0 < K; k0 += 4) {
        v2f a, bb;
        a.x  = arow[k0];
        a.y  = arow[k0 + 1];
        bb.x = wrow[k0];
        bb.y = wrow[k0 + 1];
        acc  = wmma_f32x4(a, bb, acc);
    }

    const int  colw = col0 + (lane & 15);
    const float bv  = (colw < Nreal) ? bias[colw] : 0.0f;
    #pragma unroll
    for (int i = 0; i < 8; ++i) {
        const int rowOut = row0 + i + ((lane >> 4) << 3);
        out[(size_t)rowOut * ldOut + colw] = acc[i] + bv;
    }
}

// ---------------------------------------------------------------------------
// Kernel 3: per-query combine (one wave per query n):
//   diff[s,h]   = x[h] - refx[s,h]
//   logits[s,c] = mem[refs[n,s]]·Wm[c,:] + bm[c] + sum_h diff[s,h]*grad[h,c]
//   dist        = softmax_s(-||diff_s||);  out[n,c] = sum_s dist[s]*logits[s,c]
// ---------------------------------------------------------------------------
__global__ void __launch_bounds__(32)
combine_kernel(const float* __restrict__ pos  /* NTOT x 16 */,
               const float* __restrict__ grad /* NIMG x 128 (h*10+c) */,
               const float* __restrict__ mem, const float* __restrict__ Wm,
               const float* __restrict__ bm, const int* __restrict__ refs,
               float* __restrict__ out /* NIMG x 10 */)
{
    const int n = blockIdx.x;
    const int lane = threadIdx.x;

    __shared__ float x[HID];
    __shared__ float diff[SREF][HID];
    __shared__ float logits[SREF][NCLS];
    __shared__ float negd[SREF];
    __shared__ float g[HID * NCLS];

    if (lane < HID) x[lane] = pos[(size_t)n * 16 + lane];
    for (int i = lane; i < HID * NCLS; i += 32) g[i] = grad[(size_t)n * 128 + i];
    __syncthreads();

    for (int i = lane; i < SREF * HID; i += 32) {
        const int s = i / HID, h = i % HID;
        diff[s][h] = x[h] - pos[(size_t)(NIMG + n * SREF + s) * 16 + h];
    }
    __syncthreads();

    for (int i = lane; i < SREF * NCLS; i += 32) {
        const int s = i / NCLS, c = i % NCLS;
        const int r = refs[n * SREF + s];
        float acc = bm[c];
        #pragma unroll
        for (int cc = 0; cc < NCLS; ++cc)
            acc = fmaf(mem[(size_t)r * NCLS + cc], Wm[c * NCLS + cc], acc);
        #pragma unroll
        for (int h = 0; h < HID; ++h)
            acc = fmaf(diff[s][h], g[h * NCLS + c], acc);
        logits[s][c] = acc;
    }
    if (lane < SREF) {
        float ss = 0.0f;
        #pragma unroll
        for (int h = 0; h < HID; ++h) ss = fmaf(diff[lane][h], diff[lane][h], ss);
        negd[lane] = -sqrtf(ss);
    }
    __syncthreads();

    if (lane < NCLS) {
        float mx = negd[0];
        #pragma unroll
        for (int s = 1; s < SREF; ++s) mx = fmaxf(mx, negd[s]);
        float se = 0.0f, acc = 0.0f;
        #pragma unroll
        for (int s = 0; s < SREF; ++s) {
            const float e = expf(negd[s] - mx);
            se += e;
            acc = fmaf(e, logits[s][lane], acc);
        }
        out[(size_t)n * NCLS + lane] = acc / se;
    }
}

// ---------------------------------------------------------------------------
// Host launcher. Workspace layout (floats), fully rewritten each call:
//   feat : NTOT*400 = 3,686,400   pos : NTOT*16 = 147,456
//   grad : NIMG*128 =   131,072   Wfp : 16*400  Wgp : 128*400
// Total ~16.1 MB.
// ---------------------------------------------------------------------------
extern "C" void kernel_launch(void* const* d_in, const int* in_sizes, int n_in,
                              void* d_out, int out_size, void* d_ws, size_t ws_size,
                              hipStream_t stream)
{
    const float* img   = (const float*)d_in[0];
    const float* train = (const float*)d_in[1];
    const float* mem   = (const float*)d_in[2];
    const float* w1    = (const float*)d_in[3];
    const float* b1    = (const float*)d_in[4];
    const float* w2    = (const float*)d_in[5];
    const float* b2    = (const float*)d_in[6];
    const float* Wf    = (const float*)d_in[7];
    const float* bf    = (const float*)d_in[8];
    const float* Wg    = (const float*)d_in[9];
    const float* bg    = (const float*)d_in[10];
    const float* Wm    = (const float*)d_in[11];
    const float* bm    = (const float*)d_in[12];
    /* d_in[13] = idx : unused by the reference */
    const int*   refs  = (const int*)d_in[14];
    float*       out   = (float*)d_out;

    float* feat = (float*)d_ws;
    float* pos  = feat + (size_t)NTOT * FDIM;
    float* grad = pos  + (size_t)NTOT * 16;
    float* Wfp  = grad + (size_t)NIMG * 128;
    float* Wgp  = Wfp  + (size_t)16 * FDIM;

    // 0) zero-pad FC weights (independent of kernel 1, overlaps on the stream)
    pad_weights<<<16, 256, 0, stream>>>(Wf, Wfp, HID, FDIM);
    pad_weights<<<128, 256, 0, stream>>>(Wg, Wgp, HID * NCLS, FDIM);

    // 1) LeNet features (WMMA implicit-GEMM convs) for 1024 queries + 8192 refs
    lenet_features<<<NTOT, 256, 0, stream>>>(img, train, refs, w1, b1, w2, b2, feat);

    // 2) pos = feat @ Wf^T + bf : 576 tiles = 72 blocks of 8 waves
    wmma_gemm_bias<<<72, 256, 0, stream>>>(feat, Wfp, bf, pos,
                                           NTOT, FDIM, HID, 16, 1);

    // 3) grad = feat_img @ Wg^T + bg : 64x8 tiles = 64 blocks
    wmma_gemm_bias<<<64, 256, 0, stream>>>(feat, Wgp, bg, grad,
                                           NIMG, FDIM, HID * NCLS, 128, 8);

    // 4) per-query combine -> (1024, 10)
    combine_kernel<<<NIMG, 32, 0, stream>>>(pos, grad, mem, Wm, bm, refs, out);
}